// GDN_sub_mean_26182120636488
// MI455X (gfx1250) — compile-verified
//
#include <hip/hip_runtime.h>

typedef float v2f __attribute__((ext_vector_type(2)));
typedef float v8f __attribute__((ext_vector_type(8)));

#define N_NODES 16384
#define NFEAT   64
#define NHID    64
#define BM      64     // output rows per block
#define KC      64     // K-chunk staged in LDS
#define QSZ     256    // floats per quad-K row (64 rows/cols * 4) -- no padding needed
#define LDA     68     // sub-tile row stride for the epilogue (68 % 64 == 4 -> conflict-free)

// LDS byte offset (wave-relative) of a pointer into a __shared__ array
#define LDS_OFF(p) ((unsigned)(uintptr_t)(__attribute__((address_space(3))) const void*)(p))

// CDNA5 async global->LDS copies, tracked by ASYNCcnt
static __device__ __forceinline__ void async_copy_b128(unsigned lds_byte_off, const void* gaddr) {
  asm volatile("global_load_async_to_lds_b128 %0, %1, off"
               :: "v"(lds_byte_off), "v"(gaddr) : "memory");
}
static __device__ __forceinline__ void async_copy_b32(unsigned lds_byte_off, const void* gaddr) {
  asm volatile("global_load_async_to_lds_b32 %0, %1, off"
               :: "v"(lds_byte_off), "v"(gaddr) : "memory");
}
static __device__ __forceinline__ void wait_asynccnt0() {
  asm volatile("s_wait_asynccnt 0" ::: "memory");
}

// D = A(16x4) * B(4x16) + C, fp32 WMMA, wave32
static __device__ __forceinline__ v8f wmma_f32(v2f a, v2f b, v8f c) {
  return __builtin_amdgcn_wmma_f32_16x16x4_f32(
      /*neg_a=*/false, a, /*neg_b=*/false, b,
      /*c_mod=*/(short)0, c, /*reuse_a=*/false, /*reuse_b=*/false);
}

__global__ __launch_bounds__(256, 1)
void gdn_sub_mean_fused(const float* __restrict__ xg,
                        const float* __restrict__ adj,
                        const float* __restrict__ dn,
                        const float* __restrict__ Wg,
                        const float* __restrict__ bg,
                        float* __restrict__ out) {
  // quad-K interleaved tiles: element (k, j) at [k>>2]*QSZ + j*4 + (k&3)
  __shared__ float lds_a[2][(KC / 4) * QSZ];  // adj tile, 2 x 16 KB (reused as `sub` after K loop)
  __shared__ float lds_b[2][(KC / 4) * QSZ];  // x tile,   2 x 16 KB

  const int tid   = threadIdx.x;
  const int lane  = tid & 31;
  const int wave  = tid >> 5;
  const int l15   = lane & 15;
  const int lhalf = lane >> 4;             // 0: lanes 0-15 (K,K+1), 1: lanes 16-31 (K+2,K+3)
  const int m0    = blockIdx.x * BM;

  // 4x4 grid of 16x16 C tiles over 8 waves -> 2 tiles per wave, shared A fragment
  const int mi  = wave >> 1;               // 0..3
  const int ni0 = (wave & 1) * 2;          // 0 or 2
  const int ni1 = ni0 + 1;

  // --- per-thread async-copy coordinates (fixed; only kb advances) ---
  // adj: 1024 x b128 segments per chunk -> 4/thread: segment s = tid + 256*it
  //      m = s>>4, kq = s&15 ; global (m0+m, kb+4*kq..+3) -> LDS kq*1024B + m*16B
  // x:   4096 x b32 per chunk -> 16/thread: element e = tid + 256*i
  //      k = e>>6, c = e&63  ; global (kb+k, c) -> LDS ((k>>2)*256 + c*4 + (k&3))*4B

  // ---- prologue: stage chunk 0 into buffer 0 ----
#pragma unroll
  for (int it = 0; it < 4; ++it) {
    const int s = tid + 256 * it;
    const int m = s >> 4, kq = s & 15;
    async_copy_b128(LDS_OFF(&lds_a[0][kq * QSZ + m * 4]),
                    adj + (size_t)(m0 + m) * N_NODES + kq * 4);
  }
#pragma unroll
  for (int i = 0; i < 16; ++i) {
    const int e = tid + 256 * i;
    const int k = e >> 6, c = e & 63;
    async_copy_b32(LDS_OFF(&lds_b[0][(k >> 2) * QSZ + c * 4 + (k & 3)]),
                   xg + (size_t)k * NFEAT + c);
  }
  wait_asynccnt0();
  __syncthreads();                         // chunk 0 visible to all waves

  v8f acc0 = {};
  v8f acc1 = {};

  // fragment base offsets (floats) within a buffer; k-step adds q*QSZ (immediate)
  const int aOff  = (mi * 16 + l15) * 4 + lhalf * 2;
  const int bOff0 = (ni0 * 16 + l15) * 4 + lhalf * 2;
  const int bOff1 = (ni1 * 16 + l15) * 4 + lhalf * 2;

  int buf = 0;
  // ================= main K loop: S = adj @ x =================
  for (int kb = 0; kb < N_NODES; kb += KC) {
    const bool has_next = (kb + KC) < N_NODES;
    if (has_next) {                        // stream next chunk into other buffer while WMMAs run
      const int kn = kb + KC;
      float* na = lds_a[buf ^ 1];
      float* nb = lds_b[buf ^ 1];
#pragma unroll
      for (int it = 0; it < 4; ++it) {
        const int s = tid + 256 * it;
        const int m = s >> 4, kq = s & 15;
        async_copy_b128(LDS_OFF(&na[kq * QSZ + m * 4]),
                        adj + (size_t)(m0 + m) * N_NODES + kn + kq * 4);
      }
#pragma unroll
      for (int i = 0; i < 16; ++i) {
        const int e = tid + 256 * i;
        const int k = e >> 6, c = e & 63;
        async_copy_b32(LDS_OFF(&nb[(k >> 2) * QSZ + c * 4 + (k & 3)]),
                       xg + (size_t)(kn + k) * NFEAT + c);
      }
    }

    const float* aB  = lds_a[buf] + aOff;
    const float* bB0 = lds_b[buf] + bOff0;
    const float* bB1 = lds_b[buf] + bOff1;
#pragma unroll
    for (int q = 0; q < KC / 4; ++q) {     // one quad-row = one 16x16x4 WMMA step
      const v2f a  = *(const v2f*)(aB  + q * QSZ);
      const v2f b0 = *(const v2f*)(bB0 + q * QSZ);
      const v2f b1 = *(const v2f*)(bB1 + q * QSZ);
      acc0 = wmma_f32(a, b0, acc0);
      acc1 = wmma_f32(a, b1, acc1);
    }

    wait_asynccnt0();                      // my async writes to the other buffer landed
    __syncthreads();                       // all waves done reading `buf` + all copies visible
    buf ^= 1;
  }

  // ===== epilogue 1: sub = x - degree_norm * S  -> lds_a (row-major, stride LDA) =====
  float* subt = &lds_a[0][0];              // 32 KB region, free after the K loop
  {
    const int rowb = mi * 16 + lhalf * 8;
    const int c0 = ni0 * 16 + l15;
    const int c1 = ni1 * 16 + l15;
#pragma unroll
    for (int r = 0; r < 8; ++r) {
      const int row = rowb + r;
      const float d = dn[m0 + row];
      const float* xrow = xg + (size_t)(m0 + row) * NFEAT;
      subt[row * LDA + c0] = xrow[c0] - d * acc0[r];
      subt[row * LDA + c1] = xrow[c1] - d * acc1[r];
    }
  }
  __syncthreads();

  // ============ epilogue 2: h = relu(sub @ W + b); W fragments straight from L2 ============
  v8f h0 = {};
  v8f h1 = {};
  const float* aRow2 = subt + (mi * 16 + l15) * LDA;
  {
    const int c0 = ni0 * 16 + l15;
    const int c1 = ni1 * 16 + l15;
#pragma unroll
    for (int kk = 0; kk < NFEAT; kk += 4) {
      const int kof = kk + lhalf * 2;
      const v2f a = *(const v2f*)(aRow2 + kof);
      v2f b0, b1;
      b0.x = Wg[(kof + 0) * NHID + c0];  b0.y = Wg[(kof + 1) * NHID + c0];
      b1.x = Wg[(kof + 0) * NHID + c1];  b1.y = Wg[(kof + 1) * NHID + c1];
      h0 = wmma_f32(a, b0, h0);
      h1 = wmma_f32(a, b1, h1);
    }
  }

  {
    const int rowb = mi * 16 + lhalf * 8;
    const int c0 = ni0 * 16 + l15;
    const int c1 = ni1 * 16 + l15;
    const float bias0 = bg[c0];
    const float bias1 = bg[c1];
#pragma unroll
    for (int r = 0; r < 8; ++r) {
      const int row = rowb + r;
      out[(size_t)(m0 + row) * NHID + c0] = fmaxf(h0[r] + bias0, 0.0f);
      out[(size_t)(m0 + row) * NHID + c1] = fmaxf(h1[r] + bias1, 0.0f);
    }
  }
}

extern "C" void kernel_launch(void* const* d_in, const int* in_sizes, int n_in,
                              void* d_out, int out_size, void* d_ws, size_t ws_size,
                              hipStream_t stream) {
  const float* xg  = (const float*)d_in[0];   // [N, 64]
  const float* adj = (const float*)d_in[1];   // [N, N]
  const float* dn  = (const float*)d_in[2];   // [N, 1]
  const float* Wg  = (const float*)d_in[3];   // [64, 64]
  const float* bg  = (const float*)d_in[4];   // [64]
  float* out = (float*)d_out;                 // [N, 64]

  dim3 grid(N_NODES / BM);   // 256 blocks, one 64-row stripe each
  dim3 block(256);           // 8 waves (wave32)
  gdn_sub_mean_fused<<<grid, block, 0, stream>>>(xg, adj, dn, Wg, bg, out);
}